// LRU_27247272526459
// MI455X (gfx1250) — compile-verified
//
#include <hip/hip_runtime.h>
#include <hip/hip_bf16.h>

typedef float v2f __attribute__((ext_vector_type(2)));
typedef float v8f __attribute__((ext_vector_type(8)));
typedef unsigned int u32x4 __attribute__((ext_vector_type(4)));
typedef int i32x4 __attribute__((ext_vector_type(4)));
typedef int i32x8 __attribute__((ext_vector_type(8)));

#define H_IN   128
#define H_OUT  128
#define NST    256
#define T_LEN  2048
#define BSZ    16
#define MROWS  (BSZ * T_LEN)      /* 32768 */
#define NCH    32                 /* chunks over T */
#define CL     (T_LEN / NCH)      /* 64 */

#define LDA_X  (H_IN + 4)         /* 132: padded LDS row stride, 128-dword rows */
#define LDA_S  (NST + 4)          /* 260: padded LDS row stride, 256-dword rows */
#define PADI_128 6                /* pad_interval code: 2^(6+1)=128 dwords */
#define PADI_256 7                /* pad_interval code: 2^(7+1)=256 dwords */
#define PADA_4   3                /* pad_amount code: 3 -> 4 dwords */

#define WMMA_F32(A, B, C) \
  __builtin_amdgcn_wmma_f32_16x16x4_f32(false, (A), false, (B), (short)0, (C), false, false)

__device__ __forceinline__ v2f ld2(const float* p) { return *(const v2f*)p; }

#if defined(__gfx1250__) && __has_builtin(__builtin_amdgcn_tensor_load_to_lds)
#define HAVE_TDM 1
#else
#define HAVE_TDM 0
#endif

#if HAVE_TDM
// Tensor Data Mover: 2D tile (rows x cols, f32), contiguous in global
// (gstride == cols), padded in LDS by pad_amount dwords every row.
// D# layout per CDNA5 ISA 8.3/8.4 (group0: count/lds/global/type,
// group1: data_size, pad ctl, dims, tile dims, stride).
// 6-arg builtin form (clang-23 / therock headers): (g0, g1, g2, g3, g4, cpol).
__device__ __forceinline__ void tdm_load_2d(const float* gsrc, unsigned lds_bytes,
                                            unsigned rows, unsigned cols,
                                            int pad_i_code, int pad_a_code)
{
  unsigned long long ga = (unsigned long long)(uintptr_t)gsrc;
  u32x4 g0;
  g0[0] = 1u;                                             // count=1, user desc
  g0[1] = lds_bytes;                                      // lds_addr (bytes)
  g0[2] = (unsigned)ga;                                   // global_addr[31:0]
  g0[3] = (unsigned)((ga >> 32) & 0x01FFFFFFu) | (2u << 30); // addr[56:32], type=2
  i32x8 g1;
  g1[0] = (int)((2u << 16) | (1u << 20) |                 // data_size=4B, pad_enable
                ((unsigned)pad_i_code << 22) | ((unsigned)pad_a_code << 25));
  g1[1] = (int)((cols & 0xffffu) << 16);                  // tensor_dim0[15:0]
  g1[2] = (int)((cols >> 16) | ((rows & 0xffffu) << 16)); // td0[31:16] | td1[15:0]
  g1[3] = (int)((rows >> 16) | (cols << 16));             // td1[31:16] | tile_dim0
  g1[4] = (int)(rows & 0xffffu);                          // tile_dim1 (tile_dim2=0)
  g1[5] = (int)cols;                                      // tensor_dim0_stride[31:0]
  g1[6] = 0;
  g1[7] = 0;
  i32x4 z4 = {0, 0, 0, 0};
  i32x8 z8 = {0, 0, 0, 0, 0, 0, 0, 0};
  __builtin_amdgcn_tensor_load_to_lds(g0, g1, z4, z4, z8, 0);
}
#endif

// Fallback cooperative stage (also used by host pass): gstride == cols.
__device__ __forceinline__ void stage_tile_sync(float* lds, const float* g,
                                                int rows, int cols, int lstride)
{
  for (int i = threadIdx.x; i < rows * cols; i += (int)blockDim.x) {
    int r = i / cols, c = i - r * cols;
    lds[r * lstride + c] = g[(size_t)r * cols + c];
  }
}

// ---------------------------------------------------------------------------
// Kernel 1: input projection GEMM + gamma scaling
//   Xre[m][n] = exp(gamma_log[n]) * sum_h x[m][h]*B_re[n][h]   (Xim with B_im)
// grid = 2048 blocks (one 16-row M tile, TDM-staged to LDS), 8 waves/block,
// wave covers 64 cols of the 512-wide [re|im] output (4 accumulators).
// ---------------------------------------------------------------------------
__global__ __launch_bounds__(256) void lru_proj_kernel(
    const float* __restrict__ x, const float* __restrict__ B_re,
    const float* __restrict__ B_im, const float* __restrict__ gamma_log,
    float* __restrict__ Xre, float* __restrict__ Xim)
{
  __shared__ __align__(16) float ldsA[16 * LDA_X];

  const int wave = threadIdx.x >> 5;
  const int lane = threadIdx.x & 31;
  const int half = lane >> 4;
  const int l16  = lane & 15;
  const int m0   = blockIdx.x << 4;

#if HAVE_TDM
  if (threadIdx.x < 32) {
    tdm_load_2d(x + (size_t)m0 * H_IN, (unsigned)(uintptr_t)(void*)ldsA,
                16, H_IN, PADI_128, PADA_4);
    __builtin_amdgcn_s_wait_tensorcnt(0);
  }
#else
  stage_tile_sync(ldsA, x + (size_t)m0 * H_IN, 16, H_IN, LDA_X);
#endif
  __syncthreads();

  const float* wp[4];
  float*       op[4];
  float        gam[4];
  int          ncol[4];
#pragma unroll
  for (int s = 0; s < 4; ++s) {
    int j   = wave * 64 + s * 16 + l16;          // 0..511
    int n   = (j < NST) ? j : j - NST;
    ncol[s] = n;
    wp[s]   = ((j < NST) ? B_re : B_im) + (size_t)n * H_IN + 2 * half;
    op[s]   = (j < NST) ? Xre : Xim;
    gam[s]  = expf(gamma_log[n]);
  }
  const float* ap = ldsA + l16 * LDA_X + 2 * half;

  v8f acc0 = {}, acc1 = {}, acc2 = {}, acc3 = {};
  // software-pipelined: preload k+4 before consuming k
  v2f a  = ld2(ap);
  v2f b0 = ld2(wp[0]);
  v2f b1 = ld2(wp[1]);
  v2f b2 = ld2(wp[2]);
  v2f b3 = ld2(wp[3]);
#pragma unroll 8
  for (int k = 0; k < H_IN - 4; k += 4) {
    v2f an = ld2(ap + k + 4);
    v2f c0 = ld2(wp[0] + k + 4);
    v2f c1 = ld2(wp[1] + k + 4);
    v2f c2 = ld2(wp[2] + k + 4);
    v2f c3 = ld2(wp[3] + k + 4);
    acc0 = WMMA_F32(a, b0, acc0);
    acc1 = WMMA_F32(a, b1, acc1);
    acc2 = WMMA_F32(a, b2, acc2);
    acc3 = WMMA_F32(a, b3, acc3);
    a = an; b0 = c0; b1 = c1; b2 = c2; b3 = c3;
  }
  acc0 = WMMA_F32(a, b0, acc0);
  acc1 = WMMA_F32(a, b1, acc1);
  acc2 = WMMA_F32(a, b2, acc2);
  acc3 = WMMA_F32(a, b3, acc3);

#pragma unroll
  for (int v = 0; v < 8; ++v) {
    size_t row = (size_t)(m0 + half * 8 + v) * NST;
    op[0][row + ncol[0]] = acc0[v] * gam[0];
    op[1][row + ncol[1]] = acc1[v] * gam[1];
    op[2][row + ncol[2]] = acc2[v] * gam[2];
    op[3][row + ncol[3]] = acc3[v] * gam[3];
  }
}

// ---------------------------------------------------------------------------
// Kernel 2: chunk-local scan (in place), emit per-chunk local carry.
// ---------------------------------------------------------------------------
__global__ __launch_bounds__(256) void lru_scan_local_kernel(
    float* __restrict__ Xre, float* __restrict__ Xim,
    const float* __restrict__ nu_log, const float* __restrict__ theta_log,
    float* __restrict__ car_re, float* __restrict__ car_im)
{
  const int n = threadIdx.x;
  const int b = blockIdx.x / NCH;
  const int c = blockIdx.x % NCH;
  const float lm = expf(-expf(nu_log[n]));
  const float th = expf(theta_log[n]);
  const float lr = lm * cosf(th);
  const float li = lm * sinf(th);

  float yr = 0.0f, yi = 0.0f;
  size_t base = ((size_t)b * T_LEN + (size_t)c * CL) * NST + n;
  for (int k = 0; k < CL; ++k) {
    size_t idx = base + (size_t)k * NST;
    __builtin_prefetch(&Xre[idx + 8 * NST], 0, 1);
    __builtin_prefetch(&Xim[idx + 8 * NST], 0, 1);
    float xr = Xre[idx], xi = Xim[idx];
    float nyr = lr * yr - li * yi + xr;
    float nyi = lr * yi + li * yr + xi;
    yr = nyr; yi = nyi;
    Xre[idx] = yr; Xim[idx] = yi;
  }
  size_t cidx = ((size_t)b * NCH + c) * NST + n;
  car_re[cidx] = yr;
  car_im[cidx] = yi;
}

// ---------------------------------------------------------------------------
// Kernel 3: serial scan across chunk carries with Lambda^CL; writes exclusive
// carry-in (inc) for each chunk.
// ---------------------------------------------------------------------------
__global__ __launch_bounds__(256) void lru_scan_carry_kernel(
    const float* __restrict__ nu_log, const float* __restrict__ theta_log,
    const float* __restrict__ car_re, const float* __restrict__ car_im,
    float* __restrict__ inc_re, float* __restrict__ inc_im)
{
  const int n = threadIdx.x;
  const int b = blockIdx.x;
  const float lmL = expf(-(float)CL * expf(nu_log[n]));
  const float thL = (float)CL * expf(theta_log[n]);
  const float Lr  = lmL * cosf(thL);
  const float Li  = lmL * sinf(thL);

  float ir = 0.0f, ii = 0.0f;
  for (int c = 0; c < NCH; ++c) {
    size_t idx = ((size_t)b * NCH + c) * NST + n;
    inc_re[idx] = ir;
    inc_im[idx] = ii;
    float cr = car_re[idx], ci = car_im[idx];
    float nir = Lr * ir - Li * ii + cr;
    float nii = Lr * ii + Li * ir + ci;
    ir = nir; ii = nii;
  }
}

// ---------------------------------------------------------------------------
// Kernel 4: fixup — states[t] += Lambda^{k+1} * inc[chunk], in place.
// ---------------------------------------------------------------------------
__global__ __launch_bounds__(256) void lru_scan_fix_kernel(
    float* __restrict__ Xre, float* __restrict__ Xim,
    const float* __restrict__ nu_log, const float* __restrict__ theta_log,
    const float* __restrict__ inc_re, const float* __restrict__ inc_im)
{
  const int n = threadIdx.x;
  const int b = blockIdx.x / NCH;
  const int c = blockIdx.x % NCH;
  if (c == 0) return;  // carry-in zero for first chunk (uniform per block)

  const float lm = expf(-expf(nu_log[n]));
  const float th = expf(theta_log[n]);
  const float lr = lm * cosf(th);
  const float li = lm * sinf(th);

  size_t iidx = ((size_t)b * NCH + c) * NST + n;
  float zr = inc_re[iidx], zi = inc_im[iidx];

  size_t base = ((size_t)b * T_LEN + (size_t)c * CL) * NST + n;
  for (int k = 0; k < CL; ++k) {
    float nzr = lr * zr - li * zi;
    float nzi = lr * zi + li * zr;
    zr = nzr; zi = nzi;
    size_t idx = base + (size_t)k * NST;
    __builtin_prefetch(&Xre[idx + 8 * NST], 0, 1);
    __builtin_prefetch(&Xim[idx + 8 * NST], 0, 1);
    Xre[idx] += zr;
    Xim[idx] += zi;
  }
}

// ---------------------------------------------------------------------------
// Kernel 5: output projection GEMM:
//   out[m][o] = sum_h x[m][h]*D[o][h] + sum_n (2*C_re[o][n])*Sre[m][n]
//                                     + sum_n (-2*C_im[o][n])*Sim[m][n]
// grid = 512 blocks: one 16-row M tile; A operands (x | Sre | Sim) staged to
// LDS via 3 TDM loads; 8 waves x 16 output cols, 1 accumulator each.
// ---------------------------------------------------------------------------
__global__ __launch_bounds__(256) void lru_out_kernel(
    const float* __restrict__ x, const float* __restrict__ Sre,
    const float* __restrict__ Sim, const float* __restrict__ C_re,
    const float* __restrict__ C_im, const float* __restrict__ D,
    float* __restrict__ out)
{
  __shared__ __align__(16) float ldsX[16 * LDA_X];
  __shared__ __align__(16) float ldsR[16 * LDA_S];
  __shared__ __align__(16) float ldsI[16 * LDA_S];

  const int wave = threadIdx.x >> 5;
  const int lane = threadIdx.x & 31;
  const int half = lane >> 4;
  const int l16  = lane & 15;
  const int m0   = blockIdx.x << 4;

#if HAVE_TDM
  if (threadIdx.x < 32) {
    tdm_load_2d(x   + (size_t)m0 * H_IN, (unsigned)(uintptr_t)(void*)ldsX,
                16, H_IN, PADI_128, PADA_4);
    tdm_load_2d(Sre + (size_t)m0 * NST,  (unsigned)(uintptr_t)(void*)ldsR,
                16, NST, PADI_256, PADA_4);
    tdm_load_2d(Sim + (size_t)m0 * NST,  (unsigned)(uintptr_t)(void*)ldsI,
                16, NST, PADI_256, PADA_4);
    __builtin_amdgcn_s_wait_tensorcnt(0);
  }
#else
  stage_tile_sync(ldsX, x   + (size_t)m0 * H_IN, 16, H_IN, LDA_X);
  stage_tile_sync(ldsR, Sre + (size_t)m0 * NST,  16, NST, LDA_S);
  stage_tile_sync(ldsI, Sim + (size_t)m0 * NST,  16, NST, LDA_S);
#endif
  __syncthreads();

  const int o = wave * 16 + l16;  // output column 0..127
  const float* dp  = D    + (size_t)o * H_IN + 2 * half;
  const float* crp = C_re + (size_t)o * NST  + 2 * half;
  const float* cip = C_im + (size_t)o * NST  + 2 * half;
  const float* axp = ldsX + l16 * LDA_X + 2 * half;
  const float* arp = ldsR + l16 * LDA_S + 2 * half;
  const float* aip = ldsI + l16 * LDA_S + 2 * half;

  v8f acc = {};
  {  // D part
    v2f a = ld2(axp), b = ld2(dp);
#pragma unroll 8
    for (int k = 0; k < H_IN - 4; k += 4) {
      v2f an = ld2(axp + k + 4);
      v2f bn = ld2(dp + k + 4);
      acc = WMMA_F32(a, b, acc);
      a = an; b = bn;
    }
    acc = WMMA_F32(a, b, acc);
  }
  {  // +2*C_re * Sre
    v2f a = ld2(arp), b = ld2(crp); b += b;
#pragma unroll 8
    for (int k = 0; k < NST - 4; k += 4) {
      v2f an = ld2(arp + k + 4);
      v2f bn = ld2(crp + k + 4); bn += bn;
      acc = WMMA_F32(a, b, acc);
      a = an; b = bn;
    }
    acc = WMMA_F32(a, b, acc);
  }
  {  // -2*C_im * Sim
    v2f a = ld2(aip), b = ld2(cip); b = -(b + b);
#pragma unroll 8
    for (int k = 0; k < NST - 4; k += 4) {
      v2f an = ld2(aip + k + 4);
      v2f bn = ld2(cip + k + 4); bn = -(bn + bn);
      acc = WMMA_F32(a, b, acc);
      a = an; b = bn;
    }
    acc = WMMA_F32(a, b, acc);
  }

#pragma unroll
  for (int v = 0; v < 8; ++v) {
    size_t row = (size_t)(m0 + half * 8 + v) * H_OUT;
    out[row + o] = acc[v];
  }
}

// ---------------------------------------------------------------------------
extern "C" void kernel_launch(void* const* d_in, const int* in_sizes, int n_in,
                              void* d_out, int out_size, void* d_ws, size_t ws_size,
                              hipStream_t stream) {
  const float* x         = (const float*)d_in[0];
  const float* nu_log    = (const float*)d_in[1];
  const float* theta_log = (const float*)d_in[2];
  const float* gamma_log = (const float*)d_in[3];
  const float* B_re      = (const float*)d_in[4];
  const float* B_im      = (const float*)d_in[5];
  const float* C_re      = (const float*)d_in[6];
  const float* C_im      = (const float*)d_in[7];
  const float* D         = (const float*)d_in[8];
  float* out = (float*)d_out;

  float* ws = (float*)d_ws;
  const size_t S   = (size_t)MROWS * NST;          // one complex plane: 33.5 MB
  const size_t CSZ = (size_t)BSZ * NCH * NST;      // carry plane
  float* Xre    = ws;
  float* Xim    = ws + S;
  float* car_re = ws + 2 * S;
  float* car_im = car_re + CSZ;
  float* inc_re = car_im + CSZ;
  float* inc_im = inc_re + CSZ;
  (void)in_sizes; (void)n_in; (void)out_size; (void)ws_size;

  lru_proj_kernel<<<MROWS / 16, 256, 0, stream>>>(x, B_re, B_im, gamma_log, Xre, Xim);
  lru_scan_local_kernel<<<BSZ * NCH, NST, 0, stream>>>(Xre, Xim, nu_log, theta_log, car_re, car_im);
  lru_scan_carry_kernel<<<BSZ, NST, 0, stream>>>(nu_log, theta_log, car_re, car_im, inc_re, inc_im);
  lru_scan_fix_kernel<<<BSZ * NCH, NST, 0, stream>>>(Xre, Xim, nu_log, theta_log, inc_re, inc_im);
  lru_out_kernel<<<MROWS / 64, 256, 0, stream>>>(x, Xre, Xim, C_re, C_im, D, out);
}